// GCN_730144440424
// MI455X (gfx1250) — compile-verified
//
#include <hip/hip_runtime.h>

#define N_NODES 50000
#define N_EDGES 800000
#define IN_C 256
#define HID_C 128
#define OUT_C 64

typedef __attribute__((ext_vector_type(2))) float v2f;
typedef __attribute__((ext_vector_type(8))) float v8f;

// ---------------- degree / normalization ----------------

__global__ void __launch_bounds__(256) deg_init(float* __restrict__ dis) {
  int n = blockIdx.x * blockDim.x + threadIdx.x;
  if (n < N_NODES) dis[n] = 1.0f;  // self-loop contributes 1 to every degree
}

__global__ void __launch_bounds__(256) deg_edges(const long long* __restrict__ dst,
                                                 float* __restrict__ deg) {
  int e = blockIdx.x * blockDim.x + threadIdx.x;
  if (e < N_EDGES) atomicAdd(&deg[(int)dst[e]], 1.0f);
}

__global__ void __launch_bounds__(256) deg_rsqrt(float* __restrict__ dis) {
  int n = blockIdx.x * blockDim.x + threadIdx.x;
  if (n < N_NODES) dis[n] = rsqrtf(dis[n]);  // deg >= 1 always (self-loops)
}

// ---------------- weight transpose (one-shot, tiny) ----------------
// BT[n,k] = B[k,n] so a WMMA B-fragment's two K-adjacent elements are contiguous.

__global__ void __launch_bounds__(256) transpose_w(const float* __restrict__ B,
                                                   float* __restrict__ BT,
                                                   int K, int N) {
  int i = blockIdx.x * blockDim.x + threadIdx.x;  // over K*N, coalesced reads
  if (i < K * N) {
    int k = i / N;
    int n = i - k * N;
    BT[(size_t)n * K + k] = B[i];
  }
}

// ---------------- fp32 WMMA GEMM: C[M,N] = A[M,K] @ BT^T[K,N] (+bias) -----------
// One wave owns a 16-row x N-col panel: NT 16x16 tiles share one A fragment per
// k-step -> NT independent V_WMMA_F32_16X16X4_F32 accumulation chains (hides XDL
// latency). BT is pre-transposed [N,K], so every fragment load is a b64 at a
// compile-time immediate offset off a pointer bumped by a constant.
// VGPR layouts per CDNA5 ISA 7.12.2:
//   A 16x4 : lane m=lane&15; lanes 0-15 hold K=k+0,k+1, lanes 16-31 hold K=k+2,k+3
//   B 4x16 : col n=lane&15;  vgpr j holds row K = k + j + 2*(lane>=16)
//   C 16x16: vgpr v holds row M = v + 8*(lane>=16), col N = lane&15

template <int N, int K, int NT>
__global__ void __launch_bounds__(256) wmma_gemm(
    const float* __restrict__ A, const float* __restrict__ BT,
    const float* __restrict__ bias, float* __restrict__ C, int M) {
  constexpr int TN = N / 16;   // total N tiles
  constexpr int WT = TN / NT;  // waves per 16-row panel
  const int lane = threadIdx.x & 31;
  const int wave = blockIdx.x * (blockDim.x >> 5) + (threadIdx.x >> 5);
  const int tileM = wave / WT;
  const int g     = wave - tileM * WT;  // N-tile group within the panel
  if (tileM * 16 >= M) return;  // wave-uniform: EXEC stays all-ones for WMMA

  const int half = lane >> 4;   // which half-wave
  const int l16  = lane & 15;
  const int kh   = half * 2;    // K sub-offset carried by upper lanes

  const float* __restrict__ Ap = A + (size_t)(tileM * 16 + l16) * K + kh;
  const float* __restrict__ Bp = BT + (size_t)(g * (NT * 16) + l16) * K + kh;

  v8f acc[NT];
#pragma unroll
  for (int nt = 0; nt < NT; ++nt) acc[nt] = (v8f){};

#pragma unroll 4
  for (int k = 0; k < K; k += 4) {
    v2f a;  // one b64 load, shared by all NT tiles
    a.x = Ap[0];
    a.y = Ap[1];
    v2f b[NT];
#pragma unroll
    for (int nt = 0; nt < NT; ++nt) {  // one b64 each, immediate offset nt*16*K
      b[nt].x = Bp[nt * 16 * K];
      b[nt].y = Bp[nt * 16 * K + 1];
    }
#pragma unroll
    for (int nt = 0; nt < NT; ++nt)
      acc[nt] = __builtin_amdgcn_wmma_f32_16x16x4_f32(
          false, a, false, b[nt], (short)0, acc[nt], false, false);
    Ap += 4;
    Bp += 4;
  }

  float* __restrict__ Cp =
      C + (size_t)(tileM * 16 + half * 8) * N + g * (NT * 16) + l16;
#pragma unroll
  for (int nt = 0; nt < NT; ++nt) {
    const float bv = bias ? bias[g * (NT * 16) + nt * 16 + l16] : 0.0f;
#pragma unroll
    for (int v = 0; v < 8; ++v)
      Cp[(size_t)v * N + nt * 16] = acc[nt][v] + bv;
  }
}

// ---------------- aggregation ----------------

// h[n,:] = t[n,:] * dis[n]^2   (self-loop term; also initializes h, no memset)
__global__ void __launch_bounds__(256) self_loop_init(const float4* __restrict__ t4,
                                                      const float* __restrict__ dis,
                                                      float4* __restrict__ h4) {
  int i = blockIdx.x * blockDim.x + threadIdx.x;  // over N_NODES * (HID_C/4)
  if (i < N_NODES * (HID_C / 4)) {
    float d = dis[i >> 5];  // 32 float4 per node
    float w = d * d;
    float4 v = t4[i];
    h4[i] = make_float4(v.x * w, v.y * w, v.z * w, v.w * w);
  }
}

// h[dst,:] += t[src,:] * dis[src]*dis[dst]
// One wave per edge; lane owns 4 channels: one global_load_b128 + 4 atomic_add_f32
// (h fits in the 192MB L2 -> gathers and atomics stay on-chip).
__global__ void __launch_bounds__(256) edge_scatter(const long long* __restrict__ src,
                                                    const long long* __restrict__ dst,
                                                    const float* __restrict__ dis,
                                                    const float* __restrict__ t,
                                                    float* __restrict__ h) {
  int e    = blockIdx.x * 8 + (threadIdx.x >> 5);
  int lane = threadIdx.x & 31;
  if (e < N_EDGES) {
    int s = (int)src[e];
    int d = (int)dst[e];
    float w = dis[s] * dis[d];
    float4 v = *((const float4*)(t + (size_t)s * HID_C) + lane);
    float* hd = h + (size_t)d * HID_C + lane * 4;
    atomicAdd(hd + 0, v.x * w);
    atomicAdd(hd + 1, v.y * w);
    atomicAdd(hd + 2, v.z * w);
    atomicAdd(hd + 3, v.w * w);
  }
}

__global__ void __launch_bounds__(256) bias_relu(float4* __restrict__ h4,
                                                 const float4* __restrict__ b4) {
  int i = blockIdx.x * blockDim.x + threadIdx.x;
  if (i < N_NODES * (HID_C / 4)) {
    float4 v = h4[i];
    float4 b = b4[i & (HID_C / 4 - 1)];
    v.x = fmaxf(v.x + b.x, 0.0f);
    v.y = fmaxf(v.y + b.y, 0.0f);
    v.z = fmaxf(v.z + b.z, 0.0f);
    v.w = fmaxf(v.w + b.w, 0.0f);
    h4[i] = v;
  }
}

// ---------------- host-side orchestration ----------------

extern "C" void kernel_launch(void* const* d_in, const int* in_sizes, int n_in,
                              void* d_out, int out_size, void* d_ws, size_t ws_size,
                              hipStream_t stream) {
  const float*     x      = (const float*)d_in[0];
  const long long* ei     = (const long long*)d_in[1];  // int64 (2, E)
  const float*     proj_W = (const float*)d_in[2];
  const float*     proj_b = (const float*)d_in[3];
  const float* convW[3]   = {(const float*)d_in[4], (const float*)d_in[6],
                             (const float*)d_in[8]};
  const float* convb[3]   = {(const float*)d_in[5], (const float*)d_in[7],
                             (const float*)d_in[9]};
  const float*     out_W  = (const float*)d_in[10];
  const float*     out_b  = (const float*)d_in[11];
  float*           out    = (float*)d_out;

  const long long* src = ei;            // edge_index[0]
  const long long* dst = ei + N_EDGES;  // edge_index[1]

  // workspace: dis (N f32, 256B aligned) | t (N*128 f32) | h (N*128 f32) | BT
  char*  ws  = (char*)d_ws;
  float* dis = (float*)ws;
  float* t   = (float*)(ws + ((N_NODES * sizeof(float) + 255) & ~(size_t)255));
  float* h   = t + (size_t)N_NODES * HID_C;
  float* BT  = h + (size_t)N_NODES * HID_C;  // max 256*128 floats (128 KB)

  const int elem4 = N_NODES * (HID_C / 4);

  // --- normalization coefficients ---
  deg_init <<<(N_NODES + 255) / 256, 256, 0, stream>>>(dis);
  deg_edges<<<(N_EDGES + 255) / 256, 256, 0, stream>>>(dst, dis);
  deg_rsqrt<<<(N_NODES + 255) / 256, 256, 0, stream>>>(dis);

  const int waves = N_NODES / 16;  // one wave per 16-row panel (NT spans all N)

  // --- input projection: h = x @ proj_W + proj_b ---
  transpose_w<<<(IN_C * HID_C + 255) / 256, 256, 0, stream>>>(proj_W, BT,
                                                              IN_C, HID_C);
  wmma_gemm<HID_C, IN_C, 8><<<(waves + 7) / 8, 256, 0, stream>>>(
      x, BT, proj_b, h, N_NODES);

  // --- 3 GCN layers: h = relu( scatter(norm * (h@W)[src]) + b ) ---
  for (int l = 0; l < 3; ++l) {
    transpose_w<<<(HID_C * HID_C + 255) / 256, 256, 0, stream>>>(convW[l], BT,
                                                                 HID_C, HID_C);
    wmma_gemm<HID_C, HID_C, 8><<<(waves + 7) / 8, 256, 0, stream>>>(
        h, BT, nullptr, t, N_NODES);
    self_loop_init<<<(elem4 + 255) / 256, 256, 0, stream>>>(
        (const float4*)t, dis, (float4*)h);
    edge_scatter<<<(N_EDGES + 7) / 8, 256, 0, stream>>>(src, dst, dis, t, h);
    bias_relu<<<(elem4 + 255) / 256, 256, 0, stream>>>(
        (float4*)h, (const float4*)convb[l]);
  }

  // --- output head: out = h @ out_W + out_b ---
  transpose_w<<<(HID_C * OUT_C + 255) / 256, 256, 0, stream>>>(out_W, BT,
                                                               HID_C, OUT_C);
  wmma_gemm<OUT_C, HID_C, 4><<<(waves + 7) / 8, 256, 0, stream>>>(
      h, BT, out_b, out, N_NODES);
}